// DifferentSoftQNetwork_87737591923446
// MI455X (gfx1250) — compile-verified
//
#include <hip/hip_runtime.h>

typedef __attribute__((ext_vector_type(2))) float v2f;
typedef __attribute__((ext_vector_type(8))) float v8f;

#define B_SAMP 1024
#define I_DIM  128
#define H_DIM  512
#define O_OPT  16
#define KCH    4            // layer-2 K split into 4 chunks of 128
#define KC_W   128          // chunk width

// ---------------- binning kernels ----------------
__global__ void zero_counts_kernel(int* counts) {
    if (threadIdx.x < O_OPT) counts[threadIdx.x] = 0;
}

__global__ void build_lists_kernel(const int* __restrict__ option,
                                   int* __restrict__ counts,
                                   int* __restrict__ lists) {
    int b = blockIdx.x * blockDim.x + threadIdx.x;
    if (b < B_SAMP) {
        int o = option[b] & (O_OPT - 1);
        int pos = atomicAdd(&counts[o], 1);
        lists[o * B_SAMP + pos] = b;
    }
}

// ---------------- main fused kernel ----------------
// grid: (64 row-tiles, 16 options, 4 k-chunks), block: 256 threads (8 waves)
// Block (o, t, kc) computes, for its 16 gathered rows r:
//   partial[kc][r] = sum_{k in chunk} sum_n X1[r,k] * W2[o][k,n] * W3[o][n]
// where the needed 128-column slice of X1 = S @ W1[o][:, chunk] is
// recomputed locally (cheap: 1/4 of layer-2 flops, avoids global round trip).
__global__ __launch_bounds__(256)
void moe_forward_kernel(const float* __restrict__ state,
                        const float* __restrict__ W1,
                        const float* __restrict__ W2,
                        const float* __restrict__ W3,
                        const int*  __restrict__ counts,
                        const int*  __restrict__ lists,
                        float* __restrict__ partial) {
    const int tile = blockIdx.x;
    const int o    = blockIdx.y;
    const int kc   = blockIdx.z;
    const int cnt  = counts[o];
    if (tile * 16 >= cnt) return;      // block-uniform exit, before any WMMA

    // +4 float row padding => A-fragment ds_load_b64 hits 64 distinct banks
    __shared__ float S[16][I_DIM + 4];     // gathered state rows (~8.3 KB)
    __shared__ float X1c[16][KC_W + 4];    // X1 column slice     (~8.3 KB)
    __shared__ float outacc[16];
    __shared__ int   rows[16];

    const int tid  = threadIdx.x;
    const int lane = tid & 31;
    const int wave = tid >> 5;         // 0..7
    const int m    = lane & 15;        // A row / B,D column within 16
    const int kh   = lane >> 4;        // half-wave: K offset (A,B), M offset (C/D)

    if (tid < 16) {
        int idx = tile * 16 + tid;
        rows[tid]   = (idx < cnt) ? lists[o * B_SAMP + idx] : -1;
        outacc[tid] = 0.0f;
    }
    __syncthreads();

    // gather 16 state rows (padded rows -> zeros => zero contribution)
    for (int e = tid; e < 16 * I_DIM; e += 256) {
        int r = e >> 7;
        int c = e & (I_DIM - 1);
        int gr = rows[r];
        S[r][c] = (gr >= 0) ? state[gr * I_DIM + c] : 0.0f;
    }
    __syncthreads();

    // ---------- layer 1 slice: X1c[16, 128] = S[16,128] @ W1[o][:, kc*128 ..] ----------
    // 8 column-tiles of 16, one per wave
    const float* W1o = W1 + (size_t)o * I_DIM * H_DIM;
    {
        const int nloc = wave * 16 + m;            // local column 0..127
        const int n    = kc * KC_W + nloc;         // global X1 column
        v8f c = {};
        for (int k0 = 0; k0 < I_DIM; k0 += 4) {
            const int ka = k0 + 2 * kh;
            v2f a = *(const v2f*)&S[m][ka];                 // A: (M=m, K=ka..ka+1)
            v2f b;
            b.x = W1o[(size_t)ka * H_DIM + n];              // B: (K=ka,   N=n)
            b.y = W1o[(size_t)(ka + 1) * H_DIM + n];        // B: (K=ka+1, N=n)
            c = __builtin_amdgcn_wmma_f32_16x16x4_f32(
                    false, a, false, b, (short)0, c, false, false);
        }
#pragma unroll
        for (int r = 0; r < 8; ++r)
            X1c[r + 8 * kh][nloc] = c[r];          // C: row r+8*kh, col nloc
    }
    __syncthreads();

    // ---------- layer 2 K-chunk (+ fused layer-3 dot with W3) ----------
    const float* W2o = W2 + ((size_t)o * H_DIM + (size_t)kc * KC_W) * H_DIM;
    const float* W3o = W3 + (size_t)o * H_DIM;
    v8f acc = {};
    for (int j = 0; j < 4; ++j) {
        const int n2 = (wave * 4 + j) * 16 + m;    // output column 0..511
        v8f c = {};
        for (int k0 = 0; k0 < KC_W; k0 += 4) {
            const int ka = k0 + 2 * kh;            // local k within chunk
            v2f a = *(const v2f*)&X1c[m][ka];
            v2f b;
            b.x = W2o[(size_t)ka * H_DIM + n2];
            b.y = W2o[(size_t)(ka + 1) * H_DIM + n2];
            c = __builtin_amdgcn_wmma_f32_16x16x4_f32(
                    false, a, false, b, (short)0, c, false, false);
        }
        const float w3 = W3o[n2];
#pragma unroll
        for (int r = 0; r < 8; ++r) acc[r] += c[r] * w3;   // X2[row][n2] * W3[n2]
    }

    // reduce over the 16 lanes (columns) of each half-wave; rows stay fixed
#pragma unroll
    for (int off = 8; off >= 1; off >>= 1) {
#pragma unroll
        for (int r = 0; r < 8; ++r)
            acc[r] += __shfl_xor(acc[r], off, 32);
    }
    if (m == 0) {                                   // lanes 0 and 16 of each wave
#pragma unroll
        for (int r = 0; r < 8; ++r)
            atomicAdd(&outacc[r + 8 * kh], acc[r]); // LDS ds_add_f32 (block-local)
    }
    __syncthreads();

    if (tid < 16) {
        int gr = rows[tid];
        if (gr >= 0) partial[kc * B_SAMP + gr] = outacc[tid];
    }
}

// ---------------- deterministic final reduction ----------------
__global__ void reduce_kernel(const float* __restrict__ partial,
                              float* __restrict__ out) {
    int b = blockIdx.x * blockDim.x + threadIdx.x;
    if (b < B_SAMP) {
        float s = partial[b];
#pragma unroll
        for (int c = 1; c < KCH; ++c) s += partial[c * B_SAMP + b];
        out[b] = s;
    }
}

// ---------------- launcher ----------------
extern "C" void kernel_launch(void* const* d_in, const int* in_sizes, int n_in,
                              void* d_out, int out_size, void* d_ws, size_t ws_size,
                              hipStream_t stream) {
    const float* state  = (const float*)d_in[0];
    // d_in[1] = action (unused by the reference forward)
    const float* W1     = (const float*)d_in[2];
    const float* W2     = (const float*)d_in[3];
    const float* W3     = (const float*)d_in[4];
    const int*   option = (const int*)d_in[5];
    float* out = (float*)d_out;

    int*   counts  = (int*)d_ws;                       // 16 ints (pad to 64)
    int*   lists   = counts + 64;                      // 16 * 1024 ints
    float* partial = (float*)(lists + O_OPT * B_SAMP); // 4 * 1024 floats

    zero_counts_kernel<<<1, 32, 0, stream>>>(counts);
    build_lists_kernel<<<(B_SAMP + 255) / 256, 256, 0, stream>>>(option, counts, lists);

    dim3 grid(B_SAMP / 16, O_OPT, KCH);
    moe_forward_kernel<<<grid, 256, 0, stream>>>(state, W1, W2, W3,
                                                 counts, lists, partial);

    reduce_kernel<<<(B_SAMP + 255) / 256, 256, 0, stream>>>(partial, out);
}